// CSI2PointCloudLoss_43654047597211
// MI455X (gfx1250) — compile-verified
//
#include <hip/hip_runtime.h>
#include <math.h>

typedef __attribute__((ext_vector_type(2))) float v2f;
typedef __attribute__((ext_vector_type(8))) float v8f;

#define BATCH 16
#define NPTS 4096
#define QTILES_PER_BATCH (NPTS / 16)            /* 256 */
#define TOTAL_QTILES (BATCH * QTILES_PER_BATCH) /* 4096 */
#define WAVES_PER_BLOCK 8
#define CHAM_BLOCKS (TOTAL_QTILES / WAVES_PER_BLOCK) /* 512, 32 blocks per batch */
#define DB_LDS_FLOATS (QTILES_PER_BATCH * 64)        /* 16384 floats = 64 KB */

// One wave = one 16-point query tile; loops over all db tiles of its batch.
// Per tile, ONE V_WMMA_F32_16X16X4_F32 computes  D = t^2 - 2*q.d :
//   A (16x4) = [-2qx -2qy -2qz 1]   B (4x16) = [dx dy dz t2]^T   C = inline 0
// The B operand is pre-packed in LDS in lane-consumption order during staging
// (t2 computed once per point per block), so the inner loop is just
// ds_load_b64 + wmma + min3. p^2 and the clamp are deferred past the
// min-reduction (min commutes with a per-row constant add and with max(.,0)).
__global__ __launch_bounds__(256) void chamfer_rowmin_kernel(
    const float* __restrict__ q, const float* __restrict__ db,
    float* __restrict__ blockPartial)
{
    __shared__ float sB[DB_LDS_FLOATS]; // 64 KB of 320 KB/WGP; reused for partials

    const int lane   = threadIdx.x & 31;
    const int wave   = threadIdx.x >> 5;
    const int gid    = blockIdx.x * WAVES_PER_BLOCK + wave; // query-tile id
    const int b      = gid >> 8;   // == blockIdx.x / 32 : uniform per block
    const int qtile  = gid & 255;
    const bool laneLo = lane < 16;
    const int n16    = lane & 15;

    // ---- stage db batch as WMMA-ready B operands:
    //      tile t, lane l<16 -> {dx,dy}; lane l>=16 -> {dz,t2}
    {
        const float* dbb = db + (size_t)b * NPTS * 3;
#pragma unroll
        for (int i = 0; i < NPTS / 256; ++i) {
            const int p  = threadIdx.x + i * 256; // point 0..4095
            const float dx = dbb[p * 3 + 0];
            const float dy = dbb[p * 3 + 1];
            const float dz = dbb[p * 3 + 2];
            const float t2 = dx * dx + dy * dy + dz * dz;
            float* o = sB + (p >> 4) * 64 + (p & 15) * 2;
            o[0]  = dx;
            o[1]  = dy;
            o[32] = dz;
            o[33] = t2;
        }
    }
    __syncthreads();

    // ---- query tile: A operand (x,y,z scaled by -2; K=3 slot = 1.0) ----
    const float* qb = q + (size_t)b * NPTS * 3;
    const int qrow = qtile * 16 + n16;
    const float qx = qb[qrow * 3 + 0];
    const float qy = qb[qrow * 3 + 1];
    const float qz = qb[qrow * 3 + 2];
    const float p2lane = qx * qx + qy * qy + qz * qz;

    v2f a;
    a.x = (laneLo ? qx : qz) * -2.0f;
    a.y = laneLo ? (qy * -2.0f) : 1.0f; // K=3 carries the t2 row of B

    float rowmin[8];
#pragma unroll
    for (int r = 0; r < 8; ++r) rowmin[r] = 3.4e38f;

    const float* bptr = sB + lane * 2;
    for (int t = 0; t < QTILES_PER_BATCH; ++t) {
        const v2f bm = *(const v2f*)(bptr + t * 64); // ds_load_b64

        v8f c = {}; // SRC2 = inline 0
        c = __builtin_amdgcn_wmma_f32_16x16x4_f32(
                false, a, false, bm, (short)0, c, false, false);

#pragma unroll
        for (int r = 0; r < 8; ++r)
            rowmin[r] = fminf(rowmin[r], c[r]);
    }

    // min across the 16 lanes of each half (each half owns 8 rows of the tile)
#pragma unroll
    for (int m = 1; m < 16; m <<= 1) {
#pragma unroll
        for (int r = 0; r < 8; ++r)
            rowmin[r] = fminf(rowmin[r], __shfl_xor(rowmin[r], m, 32));
    }

    // deferred: + p^2[m], clamp, sqrt
    const int moff = laneLo ? 0 : 8; // C/D: vgpr r -> row m = r + (laneHi?8:0)
    float s = 0.0f;
#pragma unroll
    for (int r = 0; r < 8; ++r) {
        const float p2v = __shfl(p2lane, r + moff, 32);
        s += sqrtf(fmaxf(p2v + rowmin[r], 0.0f));
    }
    s += __shfl_xor(s, 16, 32); // combine both halves (16 rows total)

    // reuse sB for the per-wave partials (all reads of sB are done)
    __syncthreads();
    if (lane == 0) sB[wave] = s;
    __syncthreads();
    if (threadIdx.x == 0) {
        float tot = 0.0f;
#pragma unroll
        for (int w = 0; w < WAVES_PER_BLOCK; ++w) tot += sB[w];
        blockPartial[blockIdx.x] = tot;
    }
}

// One wave = one (batch, 16x16 tile) of gram = T * T^T, K=64 via 16 WMMA steps.
__global__ __launch_bounds__(256) void reg_kernel(
    const float* __restrict__ trans, float* __restrict__ partial)
{
    const int lane = threadIdx.x & 31;
    const int wave = threadIdx.x >> 5;
    const int gid  = blockIdx.x * 8 + wave; // 0..255
    const int b    = gid >> 4;
    const int tt   = gid & 15;
    const int ti   = tt >> 2;
    const int tj   = tt & 3;
    const bool laneLo = lane < 16;
    const int n16  = lane & 15;
    const int koff = laneLo ? 0 : 2;

    const float* T    = trans + (size_t)b * 64 * 64;
    const float* arow = T + (ti * 16 + n16) * 64 + koff; // A[m][k]
    const float* brow = T + (tj * 16 + n16) * 64 + koff; // B[k][n] = T[n][k]

    v8f c = {};
    for (int k0 = 0; k0 < 64; k0 += 4) {
        v2f a, bm;
        a.x  = arow[k0];
        a.y  = arow[k0 + 1];
        bm.x = brow[k0];
        bm.y = brow[k0 + 1];
        c = __builtin_amdgcn_wmma_f32_16x16x4_f32(
                false, a, false, bm, (short)0, c, false, false);
    }

    const int j     = tj * 16 + n16;
    const int ibase = ti * 16 + (laneLo ? 0 : 8);
    float fs = 0.0f;
#pragma unroll
    for (int r = 0; r < 8; ++r) {
        const float v = c[r] - ((ibase + r == j) ? 1.0f : 0.0f);
        fs += v * v;
    }
#pragma unroll
    for (int m = 1; m < 32; m <<= 1) fs += __shfl_xor(fs, m, 32);
    if (lane == 0) partial[gid] = fs;
}

// ws layout: [0..511] pass1 block sums, [512..1023] pass2, [1024..1279] reg tiles
__global__ __launch_bounds__(256) void finalize_kernel(
    const float* __restrict__ ws, float* __restrict__ out)
{
    __shared__ float red[256];
    __shared__ float regs[16];
    const int t = threadIdx.x;

    float s = ws[t] + ws[t + 256] + ws[t + 512] + ws[t + 768];
    red[t] = s;
    __syncthreads();
    for (int stride = 128; stride > 0; stride >>= 1) {
        if (t < stride) red[t] += red[t + stride];
        __syncthreads();
    }

    if (t < 16) {
        float rb = 0.0f;
        for (int i = 0; i < 16; ++i) rb += ws[1024 + t * 16 + i];
        regs[t] = sqrtf(rb);
    }
    __syncthreads();

    if (t == 0) {
        float reg = 0.0f;
        for (int i = 0; i < 16; ++i) reg += regs[i];
        reg *= (1.0f / 16.0f);
        const float cham = red[0] * (1.0f / (float)(BATCH * NPTS));
        out[0] = cham + 0.1f * reg;
    }
}

extern "C" void kernel_launch(void* const* d_in, const int* in_sizes, int n_in,
                              void* d_out, int out_size, void* d_ws, size_t ws_size,
                              hipStream_t stream)
{
    (void)in_sizes; (void)n_in; (void)out_size; (void)ws_size;
    const float* pred  = (const float*)d_in[0];
    const float* gt    = (const float*)d_in[1];
    const float* trans = (const float*)d_in[2];
    float* ws  = (float*)d_ws;
    float* out = (float*)d_out;

    // pred -> nearest gt
    chamfer_rowmin_kernel<<<CHAM_BLOCKS, 256, 0, stream>>>(pred, gt, ws);
    // gt -> nearest pred
    chamfer_rowmin_kernel<<<CHAM_BLOCKS, 256, 0, stream>>>(gt, pred, ws + 512);
    // feature-transform regularizer
    reg_kernel<<<32, 256, 0, stream>>>(trans, ws + 1024);
    // deterministic final reduction
    finalize_kernel<<<1, 256, 0, stream>>>(ws, out);
}